// SoftCLDiceLoss_74483322847894
// MI455X (gfx1250) — compile-verified
//
#include <hip/hip_runtime.h>
#include <stdint.h>

// ---------------------------------------------------------------------------
// Soft-clDice loss for MI455X (gfx1250).
// 21 soft-erosion steps executed as 10 fused double-steps + 1 single step,
// then final centerline + deterministic reduction.
// Working set (~96 MB) stays L2-resident (192 MB L2). Tiles staged into LDS
// via CDNA5 async global->LDS loads (ASYNCcnt) + s_wait_asynccnt.
// Double-step fusion halves lab/cl L2 round-trips for the iteration phase.
// Boundary padding in the fused kernel uses quiet NaN: v_min_num/v_max_num
// (IEEE minNum/maxNum) drop NaN operands, so NaN is neutral for BOTH the min
// and max pools -- required because the intermediate plusmin tile feeds both.
// ---------------------------------------------------------------------------

#define NTHREADS  256                 // 8 wave32 waves
#define EPSF      1e-8f

constexpr int H   = 512;
constexpr int W   = 512;
constexpr int HW  = H * W;            // 1 << 18
constexpr int HW_SHIFT = 18;
constexpr int W_SHIFT  = 9;
constexpr int NCH = 32;               // 16 pred + 16 label channels

constexpr int TS      = 32;           // output tile edge
// single-step tiles
constexpr int LTS     = TS + 4;       // 36: lab tile (halo 2)
constexpr int MTS     = TS + 2;       // 34: intermediate tile (halo 1)
constexpr int LSTRIDE = 37;           // odd stride -> staggered LDS banks
// double-step tiles
constexpr int DT3     = TS + 6;       // 38: lab tile (halo 3)
constexpr int DT2     = TS + 4;       // 36: mpA tile (halo 2)
constexpr int DT1     = TS + 2;       // 34: mpB tile (halo 1)
constexpr int DSTRIDE = 39;           // odd stride

constexpr int TILESX  = W / TS;          // 16
constexpr int TILES   = TILESX * (H / TS); // 256 per channel
constexpr int RB      = 64;              // reduce blocks per channel

#ifndef __has_builtin
#define __has_builtin(x) 0
#endif

// CDNA5 async global->LDS load, one f32 per lane (GLOBAL_LOAD_ASYNC_TO_LDS_B32).
// The LDS dest VGPR holds the wave-relative LDS byte address (low 32 bits of
// the generic pointer to shared memory on amdgcn).
__device__ __forceinline__ void async_load_f32(void* lds_dst, const float* g) {
  uint32_t lo = (uint32_t)(uintptr_t)lds_dst;
  unsigned long long ga = (unsigned long long)(uintptr_t)g;
  asm volatile("global_load_async_to_lds_b32 %0, %1, off"
               :: "v"(lo), "v"(ga)
               : "memory");
}

__device__ __forceinline__ void wait_async0() {
#if __has_builtin(__builtin_amdgcn_s_wait_asynccnt)
  __builtin_amdgcn_s_wait_asynccnt(0);
#else
  asm volatile("s_wait_asynccnt 0" ::: "memory");
#endif
}

// Load an EDGE x EDGE tile (halo HALON) of one channel plane into LDS.
// In-bounds elements use the async global->LDS path; out-of-bounds positions
// get the pooling-neutral sentinel via a plain ds store.
template <int EDGE, int HALON, int STRIDE>
__device__ __forceinline__ void load_tile_t(const float* __restrict__ plane,
                                            float* tileT, int ox, int oy,
                                            float oobval) {
  for (int l = threadIdx.x; l < EDGE * EDGE; l += NTHREADS) {
    int ly = l / EDGE;                 // constant divisor -> mul/shift
    int lx = l - ly * EDGE;
    int gx = ox - HALON + lx;
    int gy = oy - HALON + ly;
    float* dst = &tileT[ly * STRIDE + lx];
    if ((unsigned)gx < (unsigned)W && (unsigned)gy < (unsigned)H) {
      async_load_f32(dst, plane + (gy << W_SHIFT) + gx);
    } else {
      *dst = oobval;
    }
  }
  wait_async0();
  __syncthreads();
}

// plus-shaped 5-point min / 3x3 max, stride s. fminf/fmaxf lower to
// v_min_num_f32 / v_max_num_f32 which drop NaN operands.
template <int S>
__device__ __forceinline__ float min5(const float* c) {
  return fminf(c[0], fminf(fminf(c[-1], c[1]), fminf(c[-S], c[S])));
}
template <int S>
__device__ __forceinline__ float max9(const float* m) {
  return fmaxf(fmaxf(fmaxf(m[-S - 1], m[-S]), fmaxf(m[-S + 1], m[-1])),
               fmaxf(fmaxf(m[0], m[1]), fmaxf(fmaxf(m[S - 1], m[S]), m[S + 1])));
}

// ------------------------------ init ---------------------------------------
// lab[0:16] = x (pred map); lab[16:32] = one_hot(y); cl = 0.
__global__ __launch_bounds__(NTHREADS)
void init_kernel(const float* __restrict__ x, const int* __restrict__ y,
                 float* __restrict__ lab, float* __restrict__ cl) {
  unsigned idx = blockIdx.x * NTHREADS + threadIdx.x;   // < 32*HW = 2^23
  int ch  = (int)(idx >> HW_SHIFT);
  int pix = (int)(idx & (HW - 1));
  float v;
  if (ch < 16) {
    v = x[idx];
  } else {
    int cc = (ch - 16) & 1;
    int b  = (ch - 16) >> 1;
    v = (y[(b << HW_SHIFT) + pix] == cc) ? 1.0f : 0.0f;
  }
  lab[idx] = v;
  cl[idx]  = 0.0f;
}

// ------------------------------ single erosion step -------------------------
// next_lab = plusmin(lab); lp = max3x3(next_lab);
// cl += relu((1-cl)*relu(lab - lp));  lab_out = next_lab.
__global__ __launch_bounds__(NTHREADS)
void cl_step_kernel(const float* __restrict__ lab_in,
                    float* __restrict__ lab_out,
                    float* __restrict__ cl) {
  extern __shared__ float smem[];
  float* labT = smem;                       // LTS x LSTRIDE
  float* mpT  = smem + LTS * LSTRIDE;       // MTS x LSTRIDE

  int ch   = blockIdx.y;
  int tile = blockIdx.x;
  int ty = tile >> 4;                       // TILESX == 16
  int tx = tile & (TILESX - 1);
  int ox = tx * TS, oy = ty * TS;
  size_t choff = (size_t)ch << HW_SHIFT;
  const float nanf = __builtin_nanf("");

  load_tile_t<LTS, 2, LSTRIDE>(lab_in + choff, labT, ox, oy, nanf);

  // stage 1: plus-min into halo-1 tile (OOB -> NaN, neutral for the max)
  for (int l = threadIdx.x; l < MTS * MTS; l += NTHREADS) {
    int my = l / MTS;
    int mx = l - my * MTS;
    int gx = ox - 1 + mx;
    int gy = oy - 1 + my;
    float v = nanf;
    if ((unsigned)gx < (unsigned)W && (unsigned)gy < (unsigned)H)
      v = min5<LSTRIDE>(&labT[(my + 1) * LSTRIDE + (mx + 1)]);
    mpT[my * LSTRIDE + mx] = v;
  }
  __syncthreads();

  // stage 2: max3x3 + pointwise update; write next_lab and cl (in place)
  for (int k = 0; k < (TS * TS) / NTHREADS; ++k) {
    int p = threadIdx.x + k * NTHREADS;
    int y = p >> 5;                        // TS == 32
    int x = p & (TS - 1);
    int gx = ox + x, gy = oy + y;
    const float* m = &mpT[(y + 1) * LSTRIDE + (x + 1)];
    float nl = m[0];
    float lp = max9<LSTRIDE>(m);
    float labv = labT[(y + 2) * LSTRIDE + (x + 2)];
    size_t gi = choff + (size_t)((gy << W_SHIFT) + gx);
    float clv = cl[gi];
    float r = fmaxf(labv - lp, 0.0f);
    clv += fmaxf((1.0f - clv) * r, 0.0f);
    cl[gi]      = clv;
    lab_out[gi] = nl;
  }
}

// ------------------------------ fused double erosion step -------------------
// Two reference loop iterations in one pass (halo 3):
//   mpA = plusmin(lab);  lpA = max3x3(mpA);  clA = cl + relu((1-cl)*relu(lab-lpA))
//   mpB = plusmin(mpA);  lpB = max3x3(mpB);  clB = clA + relu((1-clA)*relu(mpA-lpB))
//   lab_out = mpB; cl = clB.
// NaN padding makes OOB entries neutral for both the min and max pools.
__global__ __launch_bounds__(NTHREADS)
void cl_step2_kernel(const float* __restrict__ lab_in,
                     float* __restrict__ lab_out,
                     float* __restrict__ cl) {
  extern __shared__ float smem[];
  float* lab0 = smem;                              // DT3 x DSTRIDE (halo 3)
  float* mpA  = smem + DT3 * DSTRIDE;              // DT2 x DSTRIDE (halo 2)
  float* mpB  = smem + (DT3 + DT2) * DSTRIDE;      // DT1 x DSTRIDE (halo 1)

  int ch   = blockIdx.y;
  int tile = blockIdx.x;
  int ty = tile >> 4;
  int tx = tile & (TILESX - 1);
  int ox = tx * TS, oy = ty * TS;
  size_t choff = (size_t)ch << HW_SHIFT;
  const float nanf = __builtin_nanf("");

  load_tile_t<DT3, 3, DSTRIDE>(lab_in + choff, lab0, ox, oy, nanf);

  // stage 1: mpA = plusmin(lab) on halo-2 region; OOB -> NaN
  for (int l = threadIdx.x; l < DT2 * DT2; l += NTHREADS) {
    int my = l / DT2;
    int mx = l - my * DT2;
    int gx = ox - 2 + mx;
    int gy = oy - 2 + my;
    float v = nanf;
    if ((unsigned)gx < (unsigned)W && (unsigned)gy < (unsigned)H)
      v = min5<DSTRIDE>(&lab0[(my + 1) * DSTRIDE + (mx + 1)]);
    mpA[my * DSTRIDE + mx] = v;
  }
  __syncthreads();

  // stage 2a: mpB = plusmin(mpA) on halo-1 region; OOB -> NaN
  for (int l = threadIdx.x; l < DT1 * DT1; l += NTHREADS) {
    int my = l / DT1;
    int mx = l - my * DT1;
    int gx = ox - 1 + mx;
    int gy = oy - 1 + my;
    float v = nanf;
    if ((unsigned)gx < (unsigned)W && (unsigned)gy < (unsigned)H)
      v = min5<DSTRIDE>(&mpA[(my + 1) * DSTRIDE + (mx + 1)]);
    mpB[my * DSTRIDE + mx] = v;
  }

  // stage 2b (same phase; reads mpA + global cl only): first cl update
  float clA[(TS * TS) / NTHREADS];
  for (int k = 0; k < (TS * TS) / NTHREADS; ++k) {
    int p = threadIdx.x + k * NTHREADS;
    int y = p >> 5;
    int x = p & (TS - 1);
    int gx = ox + x, gy = oy + y;
    const float* m = &mpA[(y + 2) * DSTRIDE + (x + 2)];
    float lpA  = max9<DSTRIDE>(m);
    float labv = lab0[(y + 3) * DSTRIDE + (x + 3)];
    size_t gi = choff + (size_t)((gy << W_SHIFT) + gx);
    float clv = cl[gi];
    float r = fmaxf(labv - lpA, 0.0f);
    clA[k] = clv + fmaxf((1.0f - clv) * r, 0.0f);
  }
  __syncthreads();

  // stage 3: second cl update + stores
  for (int k = 0; k < (TS * TS) / NTHREADS; ++k) {
    int p = threadIdx.x + k * NTHREADS;
    int y = p >> 5;
    int x = p & (TS - 1);
    int gx = ox + x, gy = oy + y;
    const float* m = &mpB[(y + 1) * DSTRIDE + (x + 1)];
    float nl  = m[0];                        // lab after two erosions
    float lpB = max9<DSTRIDE>(m);
    float mA  = mpA[(y + 2) * DSTRIDE + (x + 2)];
    float r = fmaxf(mA - lpB, 0.0f);
    float cv = clA[k] + fmaxf((1.0f - clA[k]) * r, 0.0f);
    size_t gi = choff + (size_t)((gy << W_SHIFT) + gx);
    cl[gi]      = cv;
    lab_out[gi] = nl;
  }
}

// ------------------------------ final centerline ----------------------------
// fin = plusmin(max3x3(cl)).
__global__ __launch_bounds__(NTHREADS)
void cl_final_kernel(const float* __restrict__ cl_in,
                     float* __restrict__ fin) {
  extern __shared__ float smem[];
  float* labT = smem;
  float* mpT  = smem + LTS * LSTRIDE;

  int ch   = blockIdx.y;
  int tile = blockIdx.x;
  int ty = tile >> 4;
  int tx = tile & (TILESX - 1);
  int ox = tx * TS, oy = ty * TS;
  size_t choff = (size_t)ch << HW_SHIFT;
  const float nanf = __builtin_nanf("");

  load_tile_t<LTS, 2, LSTRIDE>(cl_in + choff, labT, ox, oy, nanf);

  // stage 1: max3x3 into halo-1 tile; OOB -> NaN (neutral for the min)
  for (int l = threadIdx.x; l < MTS * MTS; l += NTHREADS) {
    int my = l / MTS;
    int mx = l - my * MTS;
    int gx = ox - 1 + mx;
    int gy = oy - 1 + my;
    float v = nanf;
    if ((unsigned)gx < (unsigned)W && (unsigned)gy < (unsigned)H)
      v = max9<LSTRIDE>(&labT[(my + 1) * LSTRIDE + (mx + 1)]);
    mpT[my * LSTRIDE + mx] = v;
  }
  __syncthreads();

  // stage 2: plus-shaped min, write final centerline
  for (int k = 0; k < (TS * TS) / NTHREADS; ++k) {
    int p = threadIdx.x + k * NTHREADS;
    int y = p >> 5;
    int x = p & (TS - 1);
    int gx = ox + x, gy = oy + y;
    float v = min5<LSTRIDE>(&mpT[(y + 1) * LSTRIDE + (x + 1)]);
    fin[choff + (size_t)((gy << W_SHIFT) + gx)] = v;
  }
}

// ------------------------------ reductions ----------------------------------
// Per (b,c): S1=sum(clP*onehot), S2=sum(clP), S3=sum(clL*x), S4=sum(clL).
// float4/int4 streaming loads; deterministic fixed-order tree reduce.
__global__ __launch_bounds__(NTHREADS)
void reduce_partial_kernel(const float* __restrict__ finP,
                           const float* __restrict__ finL,
                           const float* __restrict__ x,
                           const int* __restrict__ y,
                           float* __restrict__ partials) {
  int ch = blockIdx.y;            // 0..15  (b = ch/2, c = ch%2)
  int b = ch >> 1, c = ch & 1;
  constexpr int PER  = HW / RB;   // 4096 (16B-aligned chunks)
  int base = blockIdx.x * PER;
  const float4* fp = (const float4*)(finP + ((size_t)ch << HW_SHIFT) + base);
  const float4* fl = (const float4*)(finL + ((size_t)ch << HW_SHIFT) + base);
  const float4* xp = (const float4*)(x    + ((size_t)ch << HW_SHIFT) + base);
  const int4*   yp = (const int4*)  (y    + ((size_t)b  << HW_SHIFT) + base);

  float s1 = 0.f, s2 = 0.f, s3 = 0.f, s4 = 0.f;
  for (int i = threadIdx.x; i < PER / 4; i += NTHREADS) {
    float4 cp = fp[i];
    float4 lb = fl[i];
    float4 xv = xp[i];
    int4   yv = yp[i];
    s1 += (yv.x == c ? cp.x : 0.f) + (yv.y == c ? cp.y : 0.f) +
          (yv.z == c ? cp.z : 0.f) + (yv.w == c ? cp.w : 0.f);
    s2 += cp.x + cp.y + cp.z + cp.w;
    s3 += lb.x * xv.x + lb.y * xv.y + lb.z * xv.z + lb.w * xv.w;
    s4 += lb.x + lb.y + lb.z + lb.w;
  }

  __shared__ float red[NTHREADS * 4];
  int tid = threadIdx.x;
  red[tid]                = s1;
  red[tid + NTHREADS]     = s2;
  red[tid + 2 * NTHREADS] = s3;
  red[tid + 3 * NTHREADS] = s4;
  __syncthreads();
  for (int s = NTHREADS / 2; s > 0; s >>= 1) {
    if (tid < s) {
      red[tid]                += red[tid + s];
      red[tid + NTHREADS]     += red[tid + NTHREADS + s];
      red[tid + 2 * NTHREADS] += red[tid + 2 * NTHREADS + s];
      red[tid + 3 * NTHREADS] += red[tid + 3 * NTHREADS + s];
    }
    __syncthreads();
  }
  if (tid == 0) {
    float* o = partials + ((size_t)(ch * RB + blockIdx.x) << 2);
    o[0] = red[0];
    o[1] = red[NTHREADS];
    o[2] = red[2 * NTHREADS];
    o[3] = red[3 * NTHREADS];
  }
}

__global__ void finalize_kernel(const float* __restrict__ partials,
                                float* __restrict__ out) {
  __shared__ float tot[64];
  __shared__ float dcs[16];
  int tid = threadIdx.x;
  if (tid < 64) {
    int ch = tid >> 2, j = tid & 3;
    float s = 0.0f;
    for (int k = 0; k < RB; ++k)
      s += partials[((ch * RB + k) << 2) + j];
    tot[tid] = s;
  }
  __syncthreads();
  if (tid < 16) {
    float S1 = tot[tid * 4 + 0], S2 = tot[tid * 4 + 1];
    float S3 = tot[tid * 4 + 2], S4 = tot[tid * 4 + 3];
    float A  = (S1 + EPSF) / (S2 + EPSF);   // cl_p_2vol_l
    float Bv = (S3 + EPSF) / (S4 + EPSF);   // cl_l_2vol_p
    dcs[tid] = 2.0f * A * Bv / (A + Bv);
  }
  __syncthreads();
  if (tid == 0) {
    float m = 0.0f;
    for (int i = 0; i < 16; ++i) m += dcs[i];
    out[0] = -(m * (1.0f / 16.0f));
  }
}

// ------------------------------ launch --------------------------------------
extern "C" void kernel_launch(void* const* d_in, const int* in_sizes, int n_in,
                              void* d_out, int out_size, void* d_ws, size_t ws_size,
                              hipStream_t stream) {
  (void)in_sizes; (void)n_in; (void)out_size; (void)ws_size;
  const float* x = (const float*)d_in[0];   // [8,2,512,512] f32
  const int*   y = (const int*)d_in[1];     // [8,512,512] i32
  float* out = (float*)d_out;               // scalar f32

  size_t mapsz = (size_t)NCH << HW_SHIFT;   // 8,388,608 floats (32 MB)
  float* ws    = (float*)d_ws;
  float* lab_a = ws;
  float* lab_b = ws + mapsz;
  float* cl    = ws + 2 * mapsz;
  float* fin   = lab_a;                     // reuse lab_a after the last step
  float* partials = ws + 3 * mapsz;         // 16*RB*4 floats

  // init: lab_a <- {x, one_hot(y)}, cl <- 0
  init_kernel<<<(unsigned)(mapsz / NTHREADS), NTHREADS, 0, stream>>>(x, y, lab_a, cl);

  dim3 sgrid((unsigned)TILES, (unsigned)NCH);
  size_t sh1 = (size_t)(LTS + MTS) * LSTRIDE * sizeof(float);        // 10360 B
  size_t sh2 = (size_t)(DT3 + DT2 + DT1) * DSTRIDE * sizeof(float);  // 16848 B

  // 21 erosion steps = 10 fused double-steps + 1 single step
  const float* cur = lab_a;
  float* nxt = lab_b;
  for (int i = 0; i < 10; ++i) {
    cl_step2_kernel<<<sgrid, NTHREADS, sh2, stream>>>(cur, nxt, cl);
    const float* t = nxt;
    nxt = (float*)cur;
    cur = t;
  }
  cl_step_kernel<<<sgrid, NTHREADS, sh1, stream>>>(cur, nxt, cl);

  cl_final_kernel<<<sgrid, NTHREADS, sh1, stream>>>(cl, fin);

  reduce_partial_kernel<<<dim3(RB, 16), NTHREADS, 0, stream>>>(
      fin, fin + ((size_t)16 << HW_SHIFT), x, y, partials);

  finalize_kernel<<<1, 64, 0, stream>>>(partials, out);
}